// GATLikeLayer_89807766159500
// MI455X (gfx1250) — compile-verified
//
#include <hip/hip_runtime.h>

#define N_NODES 50000
#define N_EDGES 800000
#define IN_FEAT 128
#define N_HEADS 4
#define OUT_FEAT 16
#define HF 64            // N_HEADS * OUT_FEAT
#define NEG_SLOPE 0.2f

typedef float v2f __attribute__((ext_vector_type(2)));
typedef float v8f __attribute__((ext_vector_type(8)));

// ---------------------------------------------------------------------------
// Kernel 0: zero the output accumulator (d_out is poisoned before timing).
// ---------------------------------------------------------------------------
__global__ __launch_bounds__(256) void gat_zero(float4* __restrict__ out, int n4) {
    int i = blockIdx.x * blockDim.x + threadIdx.x;
    if (i < n4) out[i] = float4{0.f, 0.f, 0.f, 0.f};
}

// ---------------------------------------------------------------------------
// Kernel 1: h = x @ W via V_WMMA_F32_16X16X4_F32 (full fp32 precision).
// One wave per 16-row tile; each wave produces a 16x64 tile (4 N-subtiles).
// A-matrix layout (ISA 7.12.2, 32-bit A 16x4): lane = m + 16*kh holds
//   VGPR0 = A[m][k0+2kh], VGPR1 = A[m][k0+2kh+1].
// B-matrix 4x16 mirrors C/D lane-half convention:
//   VGPR0 = B[2kh][n], VGPR1 = B[2kh+1][n], n = lane%16.
// C/D 16x16: VGPR r, lane L -> M = r + 8*(L/16), N = L%16.
// ---------------------------------------------------------------------------
__global__ __launch_bounds__(32) void gat_gemm_wmma(const float* __restrict__ x,
                                                    const float* __restrict__ W,
                                                    float* __restrict__ h) {
    const int lane = threadIdx.x & 31;
    const int m16  = lane & 15;   // row within 16-row tile / column within N-tile
    const int kh   = lane >> 4;   // K half (0 or 1)
    const int row0 = blockIdx.x * 16;

    v8f acc0 = {}, acc1 = {}, acc2 = {}, acc3 = {};
    const float* __restrict__ xrow = x + (size_t)(row0 + m16) * IN_FEAT;

#pragma unroll
    for (int k0 = 0; k0 < IN_FEAT; k0 += 4) {
        const int k = k0 + 2 * kh;
        v2f a;
        a.x = xrow[k];
        a.y = xrow[k + 1];

        const float* __restrict__ wr = W + (size_t)k * HF;   // W[k][*], next row +64
        v2f b0, b1, b2, b3;
        b0.x = wr[m16 +  0]; b0.y = wr[HF + m16 +  0];
        b1.x = wr[m16 + 16]; b1.y = wr[HF + m16 + 16];
        b2.x = wr[m16 + 32]; b2.y = wr[HF + m16 + 32];
        b3.x = wr[m16 + 48]; b3.y = wr[HF + m16 + 48];

        acc0 = __builtin_amdgcn_wmma_f32_16x16x4_f32(false, a, false, b0, (short)0, acc0, false, false);
        acc1 = __builtin_amdgcn_wmma_f32_16x16x4_f32(false, a, false, b1, (short)0, acc1, false, false);
        acc2 = __builtin_amdgcn_wmma_f32_16x16x4_f32(false, a, false, b2, (short)0, acc2, false, false);
        acc3 = __builtin_amdgcn_wmma_f32_16x16x4_f32(false, a, false, b3, (short)0, acc3, false, false);
    }

    const int crow = row0 + kh * 8;   // + r below
    const int ccol = m16;
#pragma unroll
    for (int r = 0; r < 8; ++r) {
        float* __restrict__ hp = h + (size_t)(crow + r) * HF + ccol;
        hp[ 0] = acc0[r];
        hp[16] = acc1[r];
        hp[32] = acc2[r];
        hp[48] = acc3[r];
    }
}

// ---------------------------------------------------------------------------
// Kernel 2: per-(node,head) attention logits  e_src / e_dst.
// ---------------------------------------------------------------------------
__global__ __launch_bounds__(256) void gat_scores(const float* __restrict__ h,
                                                  const float* __restrict__ a,
                                                  float* __restrict__ e_src,
                                                  float* __restrict__ e_dst) {
    int t = blockIdx.x * blockDim.x + threadIdx.x;   // t = node*N_HEADS + head
    if (t >= N_NODES * N_HEADS) return;
    const int head = t & (N_HEADS - 1);

    const float4* __restrict__ hv = (const float4*)(h + (size_t)t * OUT_FEAT);
    const float4* __restrict__ as = (const float4*)(a + head * 2 * OUT_FEAT);
    const float4* __restrict__ ad = as + (OUT_FEAT / 4);

    float s = 0.f, d = 0.f;
#pragma unroll
    for (int q = 0; q < OUT_FEAT / 4; ++q) {
        float4 hv4 = hv[q], as4 = as[q], ad4 = ad[q];
        s += hv4.x * as4.x + hv4.y * as4.y + hv4.z * as4.z + hv4.w * as4.w;
        d += hv4.x * ad4.x + hv4.y * ad4.y + hv4.z * ad4.z + hv4.w * ad4.w;
    }
    e_src[t] = s;
    e_dst[t] = d;
}

// ---------------------------------------------------------------------------
// Kernel 3: one wave32 per edge. Leaky-ReLU logits, softmax over 4 heads
// (computed redundantly per lane — just two b128 loads), then each lane
// writes 2 of the 64 message floats with hardware f32 atomics. All gathers
// and atomic bursts are lane-contiguous (128 B per half-tile).
// ---------------------------------------------------------------------------
__global__ __launch_bounds__(256) void gat_edge(const long long* __restrict__ ei,
                                                const float* __restrict__ h,
                                                const float* __restrict__ e_src,
                                                const float* __restrict__ e_dst,
                                                float* __restrict__ out) {
    const int wid  = (blockIdx.x * blockDim.x + threadIdx.x) >> 5;  // edge id
    const int lane = threadIdx.x & 31;
    if (wid >= N_EDGES) return;

    const int s = (int)ei[wid];
    const int d = (int)ei[N_EDGES + wid];

    float4 es = ((const float4*)e_src)[s];
    float4 ed = ((const float4*)e_dst)[d];

    float e0 = es.x + ed.x, e1 = es.y + ed.y, e2 = es.z + ed.z, e3 = es.w + ed.w;
    e0 = e0 > 0.f ? e0 : NEG_SLOPE * e0;
    e1 = e1 > 0.f ? e1 : NEG_SLOPE * e1;
    e2 = e2 > 0.f ? e2 : NEG_SLOPE * e2;
    e3 = e3 > 0.f ? e3 : NEG_SLOPE * e3;

    float m = fmaxf(fmaxf(e0, e1), fmaxf(e2, e3));
    float x0 = __expf(e0 - m), x1 = __expf(e1 - m), x2 = __expf(e2 - m), x3 = __expf(e3 - m);
    float inv = 1.f / (x0 + x1 + x2 + x3);
    x0 *= inv; x1 *= inv; x2 *= inv; x3 *= inv;

    const int hh = lane >> 4;                 // 0 or 1 within each 32-float half
    const float al_lo = hh ? x1 : x0;         // heads 0/1 for elements 0..31
    const float al_hi = hh ? x3 : x2;         // heads 2/3 for elements 32..63

    const float* __restrict__ hs = h + (size_t)s * HF;
    float* __restrict__ op = out + (size_t)d * HF;

    unsafeAtomicAdd(op + lane,      al_lo * hs[lane]);
    unsafeAtomicAdd(op + lane + 32, al_hi * hs[lane + 32]);
}

// ---------------------------------------------------------------------------
extern "C" void kernel_launch(void* const* d_in, const int* in_sizes, int n_in,
                              void* d_out, int out_size, void* d_ws, size_t ws_size,
                              hipStream_t stream) {
    const float*     x  = (const float*)d_in[0];
    const long long* ei = (const long long*)d_in[1];   // int64 edge_index [2, E]
    const float*     W  = (const float*)d_in[2];
    const float*     a  = (const float*)d_in[3];
    float* out = (float*)d_out;

    // Workspace carve: h [N,64] | e_src [N,4] | e_dst [N,4]
    float* h     = (float*)d_ws;
    float* e_src = h + (size_t)N_NODES * HF;
    float* e_dst = e_src + (size_t)N_NODES * N_HEADS;

    // 1) zero output accumulator (3.2M floats = 800K float4)
    {
        int n4 = (N_NODES * HF) / 4;
        gat_zero<<<(n4 + 255) / 256, 256, 0, stream>>>((float4*)out, n4);
    }
    // 2) h = x @ W  (WMMA f32 16x16x4), one wave per 16-row tile
    gat_gemm_wmma<<<N_NODES / 16, 32, 0, stream>>>(x, W, h);
    // 3) per-(node,head) logits
    {
        int nt = N_NODES * N_HEADS;
        gat_scores<<<(nt + 255) / 256, 256, 0, stream>>>(h, a, e_src, e_dst);
    }
    // 4) per-edge softmax + gather + atomic scatter (1 wave per edge)
    {
        long long threads = (long long)N_EDGES * 32;
        gat_edge<<<(int)((threads + 255) / 256), 256, 0, stream>>>(ei, h, e_src, e_dst, out);
    }
}